// Shift_and_Sum_Quantizer_77549929496786
// MI455X (gfx1250) — compile-verified
//
#include <hip/hip_runtime.h>
#include <stdint.h>

// Shift-and-sum quantizer for MI455X (gfx1250, wave32).
// Streaming, bandwidth-bound (~270MB @ 23.3TB/s => ~11.6us floor).
// Uses CDNA5 async global->LDS loads (ASYNCcnt) for an 8-deep, barrier-free,
// VGPR-free prefetch pipeline; b128 accesses; nontemporal stores.

typedef float v4f __attribute__((ext_vector_type(4)));

#define THREADS     256
#define TILE_FLOATS 1024   // 256 threads * 4 floats = 16 rows of 64 channels
#define DEPTH       8      // async pipeline stages (32 KiB LDS of 320 KiB/WGP)

__global__ __launch_bounds__(THREADS)
void ssq_kernel(const float* __restrict__ x,
                const int*   __restrict__ attn,
                float*       __restrict__ out,
                int total_tiles)
{
    __shared__ __align__(16) float buf[DEPTH][TILE_FLOATS];

    const int tid = threadIdx.x;
    const int bid = blockIdx.x;
    const int G   = gridDim.x;
    if (bid >= total_tiles) return;
    const int nt = (total_tiles - bid + G - 1) / G;   // tiles this block handles

    // Async-issue tile k (this block's k-th tile) into stage k&(DEPTH-1).
    // Each lane loads exactly the 16 bytes it will later consume itself,
    // so no workgroup barrier is needed — only this wave's ASYNCcnt.
    auto issue = [&](int k) {
        const int t = bid + k * G;
        const unsigned goff  = (unsigned)t * (TILE_FLOATS * 4u) + (unsigned)tid * 16u;
        // Flat LDS pointer: aperture lives in addr[63:32], DS byte addr in [31:0].
        const unsigned laddr =
            (unsigned)(unsigned long long)(uintptr_t)&buf[k & (DEPTH - 1)][tid * 4];
        asm volatile("global_load_async_to_lds_b128 %0, %1, %2"
                     :: "v"(laddr), "v"(goff), "s"(x) : "memory");
    };

    const int npre = nt < DEPTH ? nt : DEPTH;
    for (int j = 0; j < npre; ++j) issue(j);

    for (int k = 0; k < nt; ++k) {
        const int t   = bid + k * G;
        const int row = t * 16 + (tid >> 4);      // 16 lanes co-own one 64-ch row

        // Issue the attn load BEFORE the asynccnt wait so it overlaps the drain
        // (the asm "memory" clobber would otherwise pin it below the wait).
        const int a = attn[row];                  // uniform per 16-lane group

        // Async loads complete in order: tile k is done once the outstanding
        // count drops to (#tiles issued beyond k), capped at DEPTH-1.
        const int rem = nt - k - 1;
        const int thr = rem < (DEPTH - 1) ? rem : (DEPTH - 1);
        switch (thr) {
        case 7:  asm volatile("s_wait_asynccnt 0x7" ::: "memory"); break;
        case 6:  asm volatile("s_wait_asynccnt 0x6" ::: "memory"); break;
        case 5:  asm volatile("s_wait_asynccnt 0x5" ::: "memory"); break;
        case 4:  asm volatile("s_wait_asynccnt 0x4" ::: "memory"); break;
        case 3:  asm volatile("s_wait_asynccnt 0x3" ::: "memory"); break;
        case 2:  asm volatile("s_wait_asynccnt 0x2" ::: "memory"); break;
        case 1:  asm volatile("s_wait_asynccnt 0x1" ::: "memory"); break;
        default: asm volatile("s_wait_asynccnt 0x0" ::: "memory"); break;
        }

        const v4f v = *(const v4f*)(&buf[k & (DEPTH - 1)][tid * 4]);

        // Row min/max: local over 4 elems, then xor-shuffle within width 16.
        float mx = fmaxf(fmaxf(v.x, v.y), fmaxf(v.z, v.w));
        float mn = fminf(fminf(v.x, v.y), fminf(v.z, v.w));
        #pragma unroll
        for (int m = 8; m >= 1; m >>= 1) {
            mx = fmaxf(mx, __shfl_xor(mx, m, 16));
            mn = fminf(mn, __shfl_xor(mn, m, 16));
        }

        const float delta = fmaxf((mx - mn) / 255.0f, 1e-6f);
        const float zp    = -__builtin_rintf(mn / delta);   // RNE == jnp.round

        // Unified dither: n=1<<a rounds at offsets (2i+1-n)/(2n); n==1 -> offset 0.
        const int   n     = 1 << a;
        const float inv_n = 1.0f / (float)n;                       // exact (pow2)
        const float off0  = (1.0f - (float)n) * (0.5f * inv_n);    // exact dyadic

        float xd[4], s[4];
        const float xs[4] = {v.x, v.y, v.z, v.w};
        #pragma unroll
        for (int e = 0; e < 4; ++e) { xd[e] = xs[e] / delta; s[e] = 0.0f; }

        float off = off0;
        for (int i = 0; i < n; ++i) {               // <=8 iters, half-wave uniform
            #pragma unroll
            for (int e = 0; e < 4; ++e) s[e] += __builtin_rintf(xd[e] + off);
            off += inv_n;                           // exact dyadic steps
        }

        v4f o;
        #pragma unroll
        for (int e = 0; e < 4; ++e) {
            float xi = s[e] * inv_n + zp;           // * (1/2^a) exact
            xi = fminf(fmaxf(xi, 0.0f), 255.0f);    // clip to [0, 255]
            ((float*)&o)[e] = (xi - zp) * delta;    // dequantize
        }
        // Streaming output: don't pollute the 192MB L2 (working set > L2).
        __builtin_nontemporal_store(o, (v4f*)out + ((unsigned)t * 256u + (unsigned)tid));

        // Recycle this stage: our own ds_loads must drain before the async
        // engine may overwrite the same LDS bytes (separate counters).
        if (k + DEPTH < nt) {
            asm volatile("s_wait_dscnt 0x0" ::: "memory");
            issue(k + DEPTH);
        }
    }
}

extern "C" void kernel_launch(void* const* d_in, const int* in_sizes, int n_in,
                              void* d_out, int out_size, void* d_ws, size_t ws_size,
                              hipStream_t stream) {
    (void)n_in; (void)out_size; (void)d_ws; (void)ws_size;
    const float* x    = (const float*)d_in[0];
    const int*   attn = (const int*)d_in[1];
    float*       out  = (float*)d_out;

    const int total       = in_sizes[0];              // 4*16*8192*64 = 33554432
    const int total_tiles = total / TILE_FLOATS;      // 32768
    int grid = total_tiles < 2048 ? total_tiles : 2048;  // nt = 16 per block

    hipLaunchKernelGGL(ssq_kernel, dim3(grid), dim3(THREADS), 0, stream,
                       x, attn, out, total_tiles);
}